// GATConv_2345052143744
// MI455X (gfx1250) — compile-verified
//
#include <hip/hip_runtime.h>
#include <hip/hip_bf16.h>

#define GAT_N 100000
#define GAT_E 1600000
#define GAT_IN 128
#define GAT_H 4
#define GAT_D 32
#define NEG_SLOPE 0.2f

typedef __attribute__((ext_vector_type(2))) float v2f;
typedef __attribute__((ext_vector_type(8))) float v8f;

// ---------------------------------------------------------------------------
// Pass A: ft = feat @ W^T  using fp32 WMMA (16x16x4).
// Block = 16 nodes x 128 features; wave w handles feature tile [16w,16w+16).
// feat tile staged in LDS (row pad 132 -> conflict-free strided A reads).
// ---------------------------------------------------------------------------
__global__ void __launch_bounds__(256)
gat_gemm_wmma(const float* __restrict__ feat, const float* __restrict__ W,
              float* __restrict__ ft) {
    __shared__ float sfeat[16][132];
    const int tid  = threadIdx.x;
    const int wave = tid >> 5;
    const int lane = tid & 31;
    const int nodeBase = blockIdx.x * 16;

    // cooperative coalesced load of 16x128 fp32 feat tile (8 KB)
    for (int i = tid; i < 16 * GAT_IN; i += 256) {
        int r = i >> 7, c = i & 127;
        sfeat[r][c] = feat[(size_t)(nodeBase + r) * GAT_IN + c];
    }
    __syncthreads();

    const int fBase = wave * 16;
    const int half  = lane >> 4;    // 0: lanes 0-15, 1: lanes 16-31
    const int r     = lane & 15;
    const float* __restrict__ wrow = W + (size_t)(fBase + r) * GAT_IN;

    v8f c = {};
#pragma unroll
    for (int k0 = 0; k0 < GAT_IN; k0 += 4) {
        const int ka = k0 + 2 * half;
        // A 16x4 fp32: lane(r,half) holds M=r, K = ka, ka+1 (VGPR0, VGPR1)
        v2f a;
        a.x = sfeat[r][ka];
        a.y = sfeat[r][ka + 1];
        // B 4x16 fp32: lane(r,half) holds N=fBase+r, K = ka, ka+1
        v2f b;
        b.x = wrow[ka];
        b.y = wrow[ka + 1];
        c = __builtin_amdgcn_wmma_f32_16x16x4_f32(
            /*neg_a=*/false, a, /*neg_b=*/false, b,
            /*c_mod=*/(short)0, c, /*reuse_a=*/false, /*reuse_b=*/false);
    }

    // D 16x16 fp32: VGPR v, lane(r,half) -> element (M = v + 8*half, N = r)
#pragma unroll
    for (int v = 0; v < 8; ++v) {
        ft[(size_t)(nodeBase + v + 8 * half) * GAT_IN + fBase + r] = c[v];
    }
}

// ---------------------------------------------------------------------------
// Pass B: el/er [N,H] = sum_d ft[n,h,d] * attn_{l,r}[h,d]. Wave per (n,h).
// ---------------------------------------------------------------------------
__global__ void __launch_bounds__(256)
gat_node_logits(const float* __restrict__ ft,
                const float* __restrict__ attn_l,
                const float* __restrict__ attn_r,
                float* __restrict__ el, float* __restrict__ er) {
    const int w    = blockIdx.x * 8 + (threadIdx.x >> 5); // (node,h) index
    const int lane = threadIdx.x & 31;
    if (w >= GAT_N * GAT_H) return;
    const int node = w >> 2;
    const int h    = w & 3;
    const float v = ft[(size_t)node * GAT_IN + h * GAT_D + lane];
    float l = v * attn_l[h * GAT_D + lane];
    float r = v * attn_r[h * GAT_D + lane];
#pragma unroll
    for (int off = 16; off > 0; off >>= 1) {
        l += __shfl_xor(l, off, 32);
        r += __shfl_xor(r, off, 32);
    }
    if (lane == 0) { el[w] = l; er[w] = r; }
}

// ---------------------------------------------------------------------------
// Pass C: init out=0, emax=-inf, esum=0
// ---------------------------------------------------------------------------
__global__ void gat_init(float* __restrict__ out, float* __restrict__ emax,
                         float* __restrict__ esum) {
    const int i = blockIdx.x * 256 + threadIdx.x;
    if (i < GAT_N * GAT_IN) out[i] = 0.0f;
    if (i < GAT_N * GAT_H) {
        emax[i] = -__builtin_huge_valf();
        esum[i] = 0.0f;
    }
}

// sign-split float atomic max (emax initialized to -inf)
__device__ inline void atomicMaxF(float* addr, float val) {
    if (val >= 0.0f) atomicMax((int*)addr, __float_as_int(val));
    else             atomicMin((unsigned int*)addr, __float_as_uint(val));
}

// ---------------------------------------------------------------------------
// Pass D: e = leaky_relu(el[src] + er[dst]); segment max into emax[dst]
// ---------------------------------------------------------------------------
__global__ void __launch_bounds__(256)
gat_edge_logits(const int* __restrict__ src, const int* __restrict__ dst,
                const float* __restrict__ el, const float* __restrict__ er,
                float* __restrict__ ews, float* __restrict__ emax) {
    const int e = blockIdx.x * 256 + threadIdx.x;
    if (e >= GAT_E) return;
    const int s = src[e], d = dst[e];
    const float4 l = *(const float4*)(el + (size_t)s * 4);
    const float4 r = *(const float4*)(er + (size_t)d * 4);
    float v[4] = { l.x + r.x, l.y + r.y, l.z + r.z, l.w + r.w };
#pragma unroll
    for (int h = 0; h < 4; ++h) {
        float x = v[h];
        x = x > 0.0f ? x : NEG_SLOPE * x;
        ews[(size_t)e * 4 + h] = x;
        atomicMaxF(emax + (size_t)d * 4 + h, x);
    }
}

// ---------------------------------------------------------------------------
// Pass E: ee = exp(e - emax[dst]); segment sum into esum[dst]
// ---------------------------------------------------------------------------
__global__ void __launch_bounds__(256)
gat_edge_exp(const int* __restrict__ dst, float* __restrict__ ews,
             const float* __restrict__ emax, float* __restrict__ esum) {
    const int e = blockIdx.x * 256 + threadIdx.x;
    if (e >= GAT_E) return;
    const int d = dst[e];
#pragma unroll
    for (int h = 0; h < 4; ++h) {
        const float x = expf(ews[(size_t)e * 4 + h] - emax[(size_t)d * 4 + h]);
        ews[(size_t)e * 4 + h] = x;
        atomicAdd(esum + (size_t)d * 4 + h, x);
    }
}

// ---------------------------------------------------------------------------
// Pass F: rst[dst] += ft[src] * (ee / esum[dst]).  Wave per edge, lane = d.
// s/d/ews/esum are wave-uniform -> scalar loads; ft gather + atomics coalesced.
// ---------------------------------------------------------------------------
__global__ void __launch_bounds__(256)
gat_aggregate(const int* __restrict__ src, const int* __restrict__ dst,
              const float* __restrict__ ews, const float* __restrict__ esum,
              const float* __restrict__ ft, float* __restrict__ out) {
    const int e = blockIdx.x * 8 + (threadIdx.x >> 5);
    if (e >= GAT_E) return;
    const int lane = threadIdx.x & 31;
    const int s = src[e], d = dst[e];
#pragma unroll
    for (int h = 0; h < 4; ++h) {
        const float a = ews[(size_t)e * 4 + h] / esum[(size_t)d * 4 + h];
        atomicAdd(out + (size_t)d * GAT_IN + h * GAT_D + lane,
                  ft[(size_t)s * GAT_IN + h * GAT_D + lane] * a);
    }
}

// ---------------------------------------------------------------------------
extern "C" void kernel_launch(void* const* d_in, const int* in_sizes, int n_in,
                              void* d_out, int out_size, void* d_ws, size_t ws_size,
                              hipStream_t stream) {
    const float* feat   = (const float*)d_in[0];
    const float* W      = (const float*)d_in[1];
    const float* attn_l = (const float*)d_in[2];
    const float* attn_r = (const float*)d_in[3];
    const int*   src    = (const int*)d_in[4];
    const int*   dst    = (const int*)d_in[5];
    float* out = (float*)d_out;

    // workspace layout (floats)
    float* ws   = (float*)d_ws;
    float* ft   = ws;                                   // N*128
    float* el   = ft + (size_t)GAT_N * GAT_IN;          // N*4
    float* er   = el + (size_t)GAT_N * GAT_H;           // N*4
    float* emax = er + (size_t)GAT_N * GAT_H;           // N*4
    float* esum = emax + (size_t)GAT_N * GAT_H;         // N*4
    float* ews  = esum + (size_t)GAT_N * GAT_H;         // E*4

    // A: fp32 WMMA GEMM (N/16 = 6250 node tiles, 8 waves cover 128 features)
    gat_gemm_wmma<<<GAT_N / 16, 256, 0, stream>>>(feat, W, ft);
    // B: per-(node,head) attention dots
    gat_node_logits<<<(GAT_N * GAT_H) / 8, 256, 0, stream>>>(ft, attn_l, attn_r, el, er);
    // C: init accumulators
    gat_init<<<(GAT_N * GAT_IN + 255) / 256, 256, 0, stream>>>(out, emax, esum);
    // D: edge logits + segment max
    gat_edge_logits<<<GAT_E / 256, 256, 0, stream>>>(src, dst, el, er, ews, emax);
    // E: exp + segment sum
    gat_edge_exp<<<GAT_E / 256, 256, 0, stream>>>(dst, ews, emax, esum);
    // F: normalized weighted aggregation
    gat_aggregate<<<GAT_E / 8, 256, 0, stream>>>(src, dst, ews, esum, ft, out);
}